// DecagonModel_1142461300937
// MI455X (gfx1250) — compile-verified
//
#include <hip/hip_runtime.h>

// ---------------------------------------------------------------------------
// Decagon 2-layer hetero-GCN for MI455X (gfx1250, wave32).
//   Dense X@W via V_WMMA_F32_16X16X4_F32 (fp32 exact, ReLU fused into the
//   layer-2 A-fragment loads), SpMM via float4-vectorized per-(edge,col4)
//   global_atomic_add_f32 scatter. Workload is memory/atomic bound
//   (~250 MB @ 23.3 TB/s); WMMA covers the 3.7 GFLOP dense math for free.
// ---------------------------------------------------------------------------

typedef __attribute__((ext_vector_type(2))) float v2f;
typedef __attribute__((ext_vector_type(8))) float v8f;

#define DN0 50000
#define DN1 20000
#define DF  128
#define DH1 64
#define DH2 32
#define DE00 1600000
#define DE01 1000000
#define DE10 640000
#define DE11 640000
#define DK11 3

// ----------------------------- utility kernels -----------------------------

__global__ void zero_f32(float* __restrict__ p, long long n) {
  long long t = (long long)blockIdx.x * blockDim.x + threadIdx.x;
  if (t < n) p[t] = 0.0f;
}

// ------------------------------- WMMA GEMM ---------------------------------
// C[M x N] = act(A)[M x K] * W[K x N], all row-major fp32.
// One wave32 computes one 16x16 tile of C, looping K in steps of 4 using
// V_WMMA_F32_16X16X4_F32. RELU=true applies relu to A as it is loaded
// (fuses the layer-1 activation into the layer-2 GEMM; saves a full
// read+write pass over h0/h1).
//
// VGPR layouts (CDNA5 ISA 7.12.2):
//  A 16x4 : lanes 0-15 -> M=lane, VGPR0=K0, VGPR1=K1 ; lanes 16-31 -> K2,K3
//  B 4x16 : VGPR0: lanes0-15 row K0, lanes16-31 row K2 ; VGPR1: K1 / K3
//  C 16x16: VGPR v: lanes0-15 -> M=v, lanes16-31 -> M=v+8, N=lane&15
template <int K, int N, bool RELU>
__global__ void gemm_wmma_f32(const float* __restrict__ A,
                              const float* __restrict__ W,
                              float* __restrict__ C, int M) {
  const int wave  = (int)((blockIdx.x * (long long)blockDim.x + threadIdx.x) >> 5);
  const int lane  = threadIdx.x & 31;
  const int nTile = wave % (N / 16);
  const int mTile = wave / (N / 16);
  if (mTile * 16 >= M) return;           // wave-uniform: EXEC stays all-1s

  const int half = lane >> 4;            // 0 -> lanes 0-15, 1 -> lanes 16-31
  const int l    = lane & 15;
  const int m0   = mTile * 16;
  const int n0   = nTile * 16;

  const float* __restrict__ Arow = A + (long long)(m0 + l) * K + 2 * half;
  const float* __restrict__ Bp   = W + (long long)(2 * half) * N + n0 + l;

  v8f acc = {};
#pragma unroll 4
  for (int k = 0; k < K; k += 4) {
    v2f a = *(const v2f*)(Arow + k);     // A[m][k+2h], A[m][k+2h+1]
    if (RELU) {
      a.x = fmaxf(a.x, 0.0f);
      a.y = fmaxf(a.y, 0.0f);
    }
    v2f b;
    b.x = Bp[(long long)k * N];          // W[k+2h  ][n]
    b.y = Bp[(long long)(k + 1) * N];    // W[k+2h+1][n]
    acc = __builtin_amdgcn_wmma_f32_16x16x4_f32(
        /*neg_a=*/false, a, /*neg_b=*/false, b,
        /*c_mod=*/(short)0, acc, /*reuse_a=*/false, /*reuse_b=*/false);
  }

  float* __restrict__ Cp = C + (long long)(m0 + half * 8) * N + n0 + l;
#pragma unroll
  for (int v = 0; v < 8; ++v) Cp[(long long)v * N] = acc[v];
}

// ------------------------------ SpMM scatter -------------------------------
// out[rows[e]][c..c+3] += vals[e] * X[cols[e]][c..c+3]
// Thread t -> (edge = t / (H/4), quad q = t % (H/4), c = 4q). The H/4
// consecutive lanes of one edge read one float4 each of X (fully coalesced,
// 16B aligned since H is a multiple of 8) and issue 4 f32 atomics. 4-wide
// vectorization cuts the redundant row/col/val index loads 4x; the atomic
// count itself is irreducible for scatter-form SpMM.
template <int LOGH>
__global__ void spmm_scatter4(const int* __restrict__ rows,
                              const int* __restrict__ cols,
                              const float* __restrict__ vals,
                              const float* __restrict__ X,
                              float* __restrict__ out, int nEdges) {
  constexpr int QPE = 1 << (LOGH - 2);            // float4 quads per edge
  long long t = (long long)blockIdx.x * blockDim.x + threadIdx.x;
  int e = (int)(t >> (LOGH - 2));
  if (e >= nEdges) return;
  int q = (int)t & (QPE - 1);
  float v  = vals[e];
  int   r  = rows[e];
  int   cl = cols[e];

  const float4 x = *(const float4*)(X + ((((long long)cl) << LOGH) + 4 * q));
  float* o = out + ((((long long)r) << LOGH) + 4 * q);
  atomicAdd(o + 0, v * x.x);
  atomicAdd(o + 1, v * x.y);
  atomicAdd(o + 2, v * x.z);
  atomicAdd(o + 3, v * x.w);
}

// ------------------------------- launcher ----------------------------------

static inline int cdiv(long long a, int b) { return (int)((a + b - 1) / b); }

extern "C" void kernel_launch(void* const* d_in, const int* in_sizes, int n_in,
                              void* d_out, int out_size, void* d_ws, size_t ws_size,
                              hipStream_t stream) {
  (void)in_sizes; (void)n_in; (void)out_size; (void)ws_size;

  const float* feat0   = (const float*)d_in[0];
  const float* feat1   = (const float*)d_in[1];
  const int*   e00_row = (const int*)d_in[2];
  const int*   e00_col = (const int*)d_in[3];
  const float* e00_val = (const float*)d_in[4];
  const int*   e01_row = (const int*)d_in[5];
  const int*   e01_col = (const int*)d_in[6];
  const float* e01_val = (const float*)d_in[7];
  const int*   e10_row = (const int*)d_in[8];
  const int*   e10_col = (const int*)d_in[9];
  const float* e10_val = (const float*)d_in[10];
  const int*   e11_row = (const int*)d_in[11];   // [K11, E11] flat
  const int*   e11_col = (const int*)d_in[12];
  const float* e11_val = (const float*)d_in[13];
  const float* W1_00   = (const float*)d_in[14]; // [F,H1]
  const float* W1_01   = (const float*)d_in[15];
  const float* W1_10   = (const float*)d_in[16];
  const float* W1_11   = (const float*)d_in[17]; // [K11,F,H1]
  const float* W2_00   = (const float*)d_in[18]; // [H1,H2]
  const float* W2_01   = (const float*)d_in[19];
  const float* W2_10   = (const float*)d_in[20];
  const float* W2_11   = (const float*)d_in[21]; // [K11,H1,H2]

  float* out = (float*)d_out;                     // [N0+N1, H2]

  // ---- workspace layout (floats), 64 MB total; layer-2 reuses XW buffers --
  float* ws   = (float*)d_ws;
  float* xw00 = ws;                                   // N0*H1
  float* xw01 = xw00 + (long long)DN0 * DH1;          // N1*H1
  float* xw10 = xw01 + (long long)DN1 * DH1;          // N0*H1
  float* xw11 = xw10 + (long long)DN0 * DH1;          // K11*N1*H1
  float* h0   = xw11 + (long long)DK11 * DN1 * DH1;   // N0*H1
  float* h1   = h0   + (long long)DN0 * DH1;          // N1*H1
  float* a200 = xw00;                                 // N0*H2 (reuse)
  float* a201 = xw01;                                 // N1*H2
  float* a210 = xw10;                                 // N0*H2
  float* a211 = xw11;                                 // K11*N1*H2

  const int TB = 256;

  // ---- zero accumulators (every call: deterministic under graph replay) ---
  {
    long long nh0 = (long long)DN0 * DH1, nh1 = (long long)DN1 * DH1;
    long long nz  = (long long)(DN0 + DN1) * DH2;
    zero_f32<<<cdiv(nh0, TB), TB, 0, stream>>>(h0, nh0);
    zero_f32<<<cdiv(nh1, TB), TB, 0, stream>>>(h1, nh1);
    zero_f32<<<cdiv(nz,  TB), TB, 0, stream>>>(out, nz);
  }

  // ---- layer 1 dense: XW = feat @ W1_* (WMMA fp32) ------------------------
  {
    int w0 = (DN0 / 16) * (DH1 / 16);   // waves for an N0 x H1 product
    int w1 = (DN1 / 16) * (DH1 / 16);
    gemm_wmma_f32<DF, DH1, false><<<cdiv((long long)w0 * 32, TB), TB, 0, stream>>>(feat0, W1_00, xw00, DN0);
    gemm_wmma_f32<DF, DH1, false><<<cdiv((long long)w1 * 32, TB), TB, 0, stream>>>(feat1, W1_01, xw01, DN1);
    gemm_wmma_f32<DF, DH1, false><<<cdiv((long long)w0 * 32, TB), TB, 0, stream>>>(feat0, W1_10, xw10, DN0);
    for (int k = 0; k < DK11; ++k)
      gemm_wmma_f32<DF, DH1, false><<<cdiv((long long)w1 * 32, TB), TB, 0, stream>>>(
          feat1, W1_11 + (long long)k * DF * DH1,
          xw11 + (long long)k * DN1 * DH1, DN1);
  }

  // ---- layer 1 sparse scatter (accumulates all edge types / relations) ----
  spmm_scatter4<6><<<cdiv((long long)DE00 << 4, TB), TB, 0, stream>>>(e00_row, e00_col, e00_val, xw00, h0, DE00);
  spmm_scatter4<6><<<cdiv((long long)DE01 << 4, TB), TB, 0, stream>>>(e01_row, e01_col, e01_val, xw01, h0, DE01);
  spmm_scatter4<6><<<cdiv((long long)DE10 << 4, TB), TB, 0, stream>>>(e10_row, e10_col, e10_val, xw10, h1, DE10);
  for (int k = 0; k < DK11; ++k)
    spmm_scatter4<6><<<cdiv((long long)DE11 << 4, TB), TB, 0, stream>>>(
        e11_row + (long long)k * DE11, e11_col + (long long)k * DE11,
        e11_val + (long long)k * DE11, xw11 + (long long)k * DN1 * DH1, h1, DE11);

  // ---- layer 2 dense: relu(h) @ W2_* (ReLU fused into A-loads) ------------
  {
    int w0 = (DN0 / 16) * (DH2 / 16);
    int w1 = (DN1 / 16) * (DH2 / 16);
    gemm_wmma_f32<DH1, DH2, true><<<cdiv((long long)w0 * 32, TB), TB, 0, stream>>>(h0, W2_00, a200, DN0);
    gemm_wmma_f32<DH1, DH2, true><<<cdiv((long long)w1 * 32, TB), TB, 0, stream>>>(h1, W2_01, a201, DN1);
    gemm_wmma_f32<DH1, DH2, true><<<cdiv((long long)w0 * 32, TB), TB, 0, stream>>>(h0, W2_10, a210, DN0);
    for (int k = 0; k < DK11; ++k)
      gemm_wmma_f32<DH1, DH2, true><<<cdiv((long long)w1 * 32, TB), TB, 0, stream>>>(
          h1, W2_11 + (long long)k * DH1 * DH2,
          a211 + (long long)k * DN1 * DH2, DN1);
  }

  // ---- layer 2 sparse scatter straight into d_out -------------------------
  float* z0 = out;
  float* z1 = out + (long long)DN0 * DH2;
  spmm_scatter4<5><<<cdiv((long long)DE00 << 3, TB), TB, 0, stream>>>(e00_row, e00_col, e00_val, a200, z0, DE00);
  spmm_scatter4<5><<<cdiv((long long)DE01 << 3, TB), TB, 0, stream>>>(e01_row, e01_col, e01_val, a201, z0, DE01);
  spmm_scatter4<5><<<cdiv((long long)DE10 << 3, TB), TB, 0, stream>>>(e10_row, e10_col, e10_val, a210, z1, DE10);
  for (int k = 0; k < DK11; ++k)
    spmm_scatter4<5><<<cdiv((long long)DE11 << 3, TB), TB, 0, stream>>>(
        e11_row + (long long)k * DE11, e11_col + (long long)k * DE11,
        e11_val + (long long)k * DE11, a211 + (long long)k * DN1 * DH2, z1, DE11);
}